// GraphConvNetwork_48533130445596
// MI455X (gfx1250) — compile-verified
//
#include <hip/hip_runtime.h>

#define V_NODES 50000
#define IN_C    128
#define HID_C   128
#define OUT_C   64

typedef __attribute__((ext_vector_type(2))) float v2f;
typedef __attribute__((ext_vector_type(8))) float v8f;

// ---------------------------------------------------------------------------
// SpMM scatter: out[row[e], :] += w[e] * X[col[e], :]   (C = 128 channels)
// One wave32 per edge; lane L handles channels [4L, 4L+4) via float4 load +
// 4 hardware fp32 atomics (global_atomic_add_f32, destination lives in L2).
// ---------------------------------------------------------------------------
__global__ __launch_bounds__(256) void spmm_scatter_kernel(
    const int*   __restrict__ rows,
    const int*   __restrict__ cols,
    const float* __restrict__ ew,
    const float* __restrict__ X,
    float*       __restrict__ out,
    int E)
{
    const int wavesPerBlock = blockDim.x >> 5;
    const int e = blockIdx.x * wavesPerBlock + (threadIdx.x >> 5);
    if (e >= E) return;

    const int   lane = threadIdx.x & 31;
    const int   r    = rows[e];
    const int   c    = cols[e];
    const float wt   = ew[e];

    const float4* xs = (const float4*)(X + (size_t)c * 128);
    float*        op = out + (size_t)r * 128;

    // 128 channels / (4 per lane) == 32 lanes: exactly one float4 per lane.
    const float4 v    = xs[lane];
    const int    base = lane * 4;
    unsafeAtomicAdd(op + base + 0, wt * v.x);
    unsafeAtomicAdd(op + base + 1, wt * v.y);
    unsafeAtomicAdd(op + base + 2, wt * v.z);
    unsafeAtomicAdd(op + base + 3, wt * v.w);
}

// ---------------------------------------------------------------------------
// FP32 WMMA GEMM + bias (+ optional ReLU):  Out[M,N] = act(A[M,128] @ W[128,N] + b)
// One wave32 per 16x16 output tile, K swept 4 at a time with
// V_WMMA_F32_16X16X4_F32 (exact fp32 accumulation, matches reference math).
//
// Operand layouts per CDNA5 ISA 7.12.2:
//   A 16x4 f32 : lane (l&15) = row M; (l>>4) selects K pair {0,1} vs {2,3};
//                v2f = A[M][kb + 2*(l>>4) + {0,1}]
//   B 4x16 f32 : lane (l&15) = col N; (l>>4) selects K pair;
//                v2f = W[kb + 2*(l>>4) + {0,1}][N]
//   C/D 16x16  : v8f, VGPR r -> rows {r, r+8} selected by (l>>4), col = (l&15)
// ---------------------------------------------------------------------------
template<int N, bool RELU>
__global__ __launch_bounds__(32) void wmma_gemm_bias_kernel(
    const float* __restrict__ A,
    const float* __restrict__ W,
    const float* __restrict__ bias,
    float*       __restrict__ Out)
{
    const int lane = threadIdx.x & 31;
    const int rm   = lane & 15;   // row (A) / col (B) within tile
    const int kh   = lane >> 4;   // selects K pair within the 4-wide slab
    const int m0   = blockIdx.x * 16;
    const int n0   = blockIdx.y * 16;

    const float* arow = A + (size_t)(m0 + rm) * 128 + kh * 2;  // K stride 1
    const float* bcol = W + (size_t)(kh * 2) * N + (n0 + rm);  // K stride N

    v8f acc = {};
    #pragma unroll
    for (int kb = 0; kb < 128; kb += 4) {
        v2f a, b;
        a.x = arow[kb + 0];
        a.y = arow[kb + 1];
        b.x = bcol[(size_t)(kb + 0) * N];
        b.y = bcol[(size_t)(kb + 1) * N];
        // 8 args: (neg_a, A, neg_b, B, c_mod, C, reuse_a, reuse_b)
        acc = __builtin_amdgcn_wmma_f32_16x16x4_f32(
            false, a, false, b, (short)0, acc, false, false);
    }

    const float bv = bias[n0 + rm];
    #pragma unroll
    for (int r = 0; r < 8; ++r) {
        const int m   = m0 + r + kh * 8;
        float     val = acc[r] + bv;
        if (RELU) val = fmaxf(val, 0.0f);
        Out[(size_t)m * N + (n0 + rm)] = val;
    }
}

// ---------------------------------------------------------------------------
// Orchestration:
//   agg  = A @ X          (scatter atomics)
//   h    = relu(agg@W1+b1)  (fp32 WMMA)
//   agg  = A @ h          (re-zero + scatter)
//   out  = agg@W2+b2        (fp32 WMMA)
// ---------------------------------------------------------------------------
extern "C" void kernel_launch(void* const* d_in, const int* in_sizes, int n_in,
                              void* d_out, int out_size, void* d_ws, size_t ws_size,
                              hipStream_t stream)
{
    const float* X    = (const float*)d_in[0];
    const float* W1   = (const float*)d_in[1];
    const float* b1   = (const float*)d_in[2];
    const float* W2   = (const float*)d_in[3];
    const float* b2   = (const float*)d_in[4];
    const int*   erow = (const int*)d_in[5];
    const int*   ecol = (const int*)d_in[6];
    const float* ew   = (const float*)d_in[7];
    float*       out  = (float*)d_out;

    const int E = in_sizes[7];   // edge_w element count

    const size_t aggBytes = (size_t)V_NODES * 128 * sizeof(float);  // 25.6 MB
    float* agg = (float*)d_ws;                         // spmm accumulator (reused)
    float* h   = (float*)((char*)d_ws + aggBytes);     // hidden activations

    const int wavesPerBlock = 8;                       // 256 threads
    const int spmmBlocks    = (E + wavesPerBlock - 1) / wavesPerBlock;

    // ---- layer 1 ----
    hipMemsetAsync(agg, 0, aggBytes, stream);
    spmm_scatter_kernel<<<spmmBlocks, 256, 0, stream>>>(erow, ecol, ew, X, agg, E);

    {
        dim3 grid(V_NODES / 16, HID_C / 16);           // 3125 x 8 tiles
        wmma_gemm_bias_kernel<HID_C, true><<<grid, 32, 0, stream>>>(agg, W1, b1, h);
    }

    // ---- layer 2 ----
    hipMemsetAsync(agg, 0, aggBytes, stream);
    spmm_scatter_kernel<<<spmmBlocks, 256, 0, stream>>>(erow, ecol, ew, h, agg, E);

    {
        dim3 grid(V_NODES / 16, OUT_C / 16);           // 3125 x 4 tiles
        wmma_gemm_bias_kernel<OUT_C, false><<<grid, 32, 0, stream>>>(agg, W2, b2, out);
    }
}